// SpectralFilterBank_81217831568099
// MI455X (gfx1250) — compile-verified
//
#include <hip/hip_runtime.h>
#include <cstdint>

#define C128 128

typedef __attribute__((ext_vector_type(2))) float v2f;
typedef __attribute__((ext_vector_type(8))) float v8f;

// ---------------- utility kernels ----------------

__global__ void fill_kernel(float* __restrict__ p, float v, size_t n) {
  size_t i = (size_t)blockIdx.x * blockDim.x + threadIdx.x;
  size_t stride = (size_t)gridDim.x * blockDim.x;
  for (; i < n; i += stride) p[i] = v;
}

__global__ void neg_init_kernel(float* __restrict__ y, const float* __restrict__ s, size_t n) {
  size_t i = (size_t)blockIdx.x * blockDim.x + threadIdx.x;
  size_t stride = (size_t)gridDim.x * blockDim.x;
  for (; i < n; i += stride) y[i] = -s[i];
}

__global__ void degree_kernel(float* __restrict__ deg, const long long* __restrict__ row, int E) {
  int e = blockIdx.x * blockDim.x + threadIdx.x;
  if (e < E) {
    __hip_atomic_fetch_add(&deg[row[e]], 1.0f, __ATOMIC_RELAXED, __HIP_MEMORY_SCOPE_AGENT);
  }
}

// in-place deg -> dinv
__global__ void dinv_kernel(float* __restrict__ deg, int N) {
  int v = blockIdx.x * blockDim.x + threadIdx.x;
  if (v < N) {
    float d = deg[v];
    deg[v] = (d > 0.0f) ? rsqrtf(fmaxf(d, 1e-12f)) : 0.0f;
  }
}

// lw[e] = -dinv[row]*dinv[col] - 1   (self loops have weight 0 and are dropped)
__global__ void edgew_kernel(float* __restrict__ lw, const float* __restrict__ dinv,
                             const long long* __restrict__ row, const long long* __restrict__ col,
                             int E) {
  int e = blockIdx.x * blockDim.x + threadIdx.x;
  if (e < E) {
    lw[e] = -dinv[row[e]] * dinv[col[e]] - 1.0f;
  }
}

// y[dst] += scale * lw[e] * x[src]   (y pre-initialized to 0 or -Tprev)
// one wave (32 lanes) per edge; each lane handles 4 channels (float4)
__global__ void prop_kernel(float* __restrict__ y, const float* __restrict__ xin,
                            const long long* __restrict__ src, const long long* __restrict__ dst,
                            const float* __restrict__ lw, float scale, int E) {
  long long t = (long long)blockIdx.x * blockDim.x + threadIdx.x;
  int e = (int)(t >> 5);
  if (e >= E) return;
  int lane = (int)(t & 31);
  long long s = src[e];
  long long d = dst[e];
  float w = lw[e] * scale;
  const float4 v = *(const float4*)(xin + s * C128 + lane * 4);
  float* o = y + d * C128 + lane * 4;
  __hip_atomic_fetch_add(o + 0, w * v.x, __ATOMIC_RELAXED, __HIP_MEMORY_SCOPE_AGENT);
  __hip_atomic_fetch_add(o + 1, w * v.y, __ATOMIC_RELAXED, __HIP_MEMORY_SCOPE_AGENT);
  __hip_atomic_fetch_add(o + 2, w * v.z, __ATOMIC_RELAXED, __HIP_MEMORY_SCOPE_AGENT);
  __hip_atomic_fetch_add(o + 3, w * v.w, __ATOMIC_RELAXED, __HIP_MEMORY_SCOPE_AGENT);
}

// ---------------- WMMA GEMM accumulate ----------------
// outs[f] += T @ cheb_w[off[f]+k]  for f = lo..3
// block = 256 threads (8 waves); wave w -> 16-col tile w; blockIdx.x -> 16-row tile;
// blockIdx.y -> filter (lo + y). Uses V_WMMA_F32_16X16X4_F32, chaining C across k.
__global__ __launch_bounds__(256) void gemm_acc_kernel(
    const float* __restrict__ T, const float* __restrict__ cheb_w,
    float* __restrict__ outs, int k, int lo, int N) {
  const int wave = threadIdx.x >> 5;
  const int lane = threadIdx.x & 31;
  const int half = lane >> 4;      // 0: lanes 0-15, 1: lanes 16-31
  const int lr   = lane & 15;
  const int rowBase = blockIdx.x * 16;
  const int colBase = wave * 16;
  const int f = lo + blockIdx.y;

  const int offTab[4] = {0, 5, 15, 30};
  const float* __restrict__ W = cheb_w + (size_t)(offTab[f] + k) * C128 * C128;
  float* __restrict__ out = outs + (size_t)f * N * C128;

  // C/D layout (32-bit 16x16): VGPR r, lanes 0-15 -> M=r, N=lane; lanes 16-31 -> M=r+8, N=lane-16
  v8f c;
#pragma unroll
  for (int r = 0; r < 8; ++r)
    c[r] = out[(size_t)(rowBase + half * 8 + r) * C128 + colBase + lr];

  // A (16x4 f32): lane<16 holds M=lr, {K=k0, k0+1}; lane>=16 holds M=lr, {K=k0+2, k0+3}
  const float* __restrict__ arow = T + (size_t)(rowBase + lr) * C128 + half * 2;
  // B (4x16 f32): lane<16 holds N=lr, rows {k0, k0+1}; lane>=16 rows {k0+2, k0+3}
  const float* __restrict__ bcol = W + (size_t)(half * 2) * C128 + colBase + lr;

#pragma unroll
  for (int k0 = 0; k0 < C128; k0 += 4) {
    v2f a = *(const v2f*)(arow + k0);
    v2f b;
    b.x = bcol[(size_t)k0 * C128];
    b.y = bcol[(size_t)k0 * C128 + C128];
    c = __builtin_amdgcn_wmma_f32_16x16x4_f32(false, a, false, b, (short)0, c, false, false);
  }

#pragma unroll
  for (int r = 0; r < 8; ++r)
    out[(size_t)(rowBase + half * 8 + r) * C128 + colBase + lr] = c[r];
}

// ---------------- attention + combine + layernorm ----------------
// one block of 128 threads per node
__global__ __launch_bounds__(128) void finalize_kernel(
    const float* __restrict__ outs, const float* __restrict__ cheb_b,
    const float* __restrict__ attn_w, const float* __restrict__ attn_b,
    const float* __restrict__ ln_g, const float* __restrict__ ln_b,
    float* __restrict__ out, int N) {
  const int n = blockIdx.x;
  const int c = threadIdx.x;  // 0..127
  __shared__ float sbuf[128];
  __shared__ float slog[4];

  float o[4];
#pragma unroll
  for (int i = 0; i < 4; ++i)
    o[i] = outs[((size_t)i * N + n) * C128 + c] + cheb_b[i * C128 + c];

  // attention logits: logit[j] = sum_{i,c} o_i[c] * attn_w[(i*128+c), j] + attn_b[j]
#pragma unroll
  for (int j = 0; j < 4; ++j) {
    float p = 0.0f;
#pragma unroll
    for (int i = 0; i < 4; ++i)
      p += o[i] * attn_w[(size_t)(i * C128 + c) * 4 + j];
    __syncthreads();
    sbuf[c] = p;
    __syncthreads();
    for (int s = 64; s > 0; s >>= 1) {
      if (c < s) sbuf[c] += sbuf[c + s];
      __syncthreads();
    }
    if (c == 0) slog[j] = sbuf[0] + attn_b[j];
  }
  __syncthreads();

  float m = fmaxf(fmaxf(slog[0], slog[1]), fmaxf(slog[2], slog[3]));
  float e[4];
  float es = 0.0f;
#pragma unroll
  for (int j = 0; j < 4; ++j) { e[j] = expf(slog[j] - m); es += e[j]; }
  float inv = 1.0f / es;
  float cmb = 0.0f;
#pragma unroll
  for (int j = 0; j < 4; ++j) cmb += e[j] * inv * o[j];

  // mean
  __syncthreads();
  sbuf[c] = cmb;
  __syncthreads();
  for (int s = 64; s > 0; s >>= 1) {
    if (c < s) sbuf[c] += sbuf[c + s];
    __syncthreads();
  }
  float mu = sbuf[0] * (1.0f / C128);
  __syncthreads();
  // variance
  float d = cmb - mu;
  sbuf[c] = d * d;
  __syncthreads();
  for (int s = 64; s > 0; s >>= 1) {
    if (c < s) sbuf[c] += sbuf[c + s];
    __syncthreads();
  }
  float var = sbuf[0] * (1.0f / C128);

  out[(size_t)n * C128 + c] = d * rsqrtf(var + 1e-5f) * ln_g[c] + ln_b[c];
}

// ---------------- host orchestration ----------------

extern "C" void kernel_launch(void* const* d_in, const int* in_sizes, int n_in,
                              void* d_out, int out_size, void* d_ws, size_t ws_size,
                              hipStream_t stream) {
  const float*     x      = (const float*)d_in[0];
  const long long* ei     = (const long long*)d_in[1];
  const float*     cheb_w = (const float*)d_in[2];
  const float*     cheb_b = (const float*)d_in[3];
  const float*     attn_w = (const float*)d_in[4];
  const float*     attn_b = (const float*)d_in[5];
  const float*     ln_g   = (const float*)d_in[6];
  const float*     ln_b   = (const float*)d_in[7];
  float*           out    = (float*)d_out;

  const int N = in_sizes[0] / C128;   // 50000 (divisible by 16)
  const int E = in_sizes[1] / 2;      // 800000

  // workspace layout (floats): deg[N] | lw[E] | B0 | B1 | B2 | outs[4*N*128]
  float* ws   = (float*)d_ws;
  float* deg  = ws;
  float* lw   = deg + N;
  float* B0   = lw + E;
  float* B1   = B0 + (size_t)N * C128;
  float* B2   = B1 + (size_t)N * C128;
  float* outs = B2 + (size_t)N * C128;

  const long long* row = ei;       // src
  const long long* col = ei + E;   // dst

  // Laplacian edge weights
  fill_kernel<<<256, 256, 0, stream>>>(deg, 0.0f, (size_t)N);
  degree_kernel<<<(E + 255) / 256, 256, 0, stream>>>(deg, row, E);
  dinv_kernel<<<(N + 255) / 256, 256, 0, stream>>>(deg, N);
  edgew_kernel<<<(E + 255) / 256, 256, 0, stream>>>(lw, deg, row, col, E);

  fill_kernel<<<2048, 256, 0, stream>>>(outs, 0.0f, (size_t)4 * N * C128);

  const int rowTiles = N / 16;
  const long long propThreads = (long long)E * 32;
  const unsigned propBlocks = (unsigned)((propThreads + 255) / 256);

  // k = 0 : T0 = x
  gemm_acc_kernel<<<dim3(rowTiles, 4), 256, 0, stream>>>(x, cheb_w, outs, 0, 0, N);

  // k = 1 : T1 = P x
  fill_kernel<<<2048, 256, 0, stream>>>(B0, 0.0f, (size_t)N * C128);
  prop_kernel<<<propBlocks, 256, 0, stream>>>(B0, x, row, col, lw, 1.0f, E);
  gemm_acc_kernel<<<dim3(rowTiles, 4), 256, 0, stream>>>(B0, cheb_w, outs, 1, 0, N);

  // k >= 2 : Tk = 2*P*T(k-1) - T(k-2), rolling buffers
  const float* Tprev = x;
  const float* Tcur  = B0;
  float* cyc[3] = {B1, B2, B0};
  int fi = 0;
  for (int k = 2; k < 20; ++k) {
    float* Tnew = cyc[fi];
    fi = (fi + 1) % 3;
    const int lo = (k < 5) ? 0 : (k < 10) ? 1 : (k < 15) ? 2 : 3;
    neg_init_kernel<<<2048, 256, 0, stream>>>(Tnew, Tprev, (size_t)N * C128);
    prop_kernel<<<propBlocks, 256, 0, stream>>>(Tnew, Tcur, row, col, lw, 2.0f, E);
    gemm_acc_kernel<<<dim3(rowTiles, 4 - lo), 256, 0, stream>>>(Tnew, cheb_w, outs, k, lo, N);
    Tprev = Tcur;
    Tcur  = Tnew;
  }

  finalize_kernel<<<N, 128, 0, stream>>>(outs, cheb_b, attn_w, attn_b, ln_g, ln_b, out, N);
}